// SineGenVITS_17248588660875
// MI455X (gfx1250) — compile-verified
//
#include <hip/hip_runtime.h>
#include <math.h>

// SineGen (NSF/VITS harmonic source) for MI455X gfx1250.
// Memory-bound: ~243 MB total traffic -> ~10.4us floor at 23.3 TB/s.
// Intra-chunk prefix-scan via V_WMMA_F32_16X16X4_F32 (scan == lower-
// triangular-ones matmul); f0 staged to LDS via gfx1250 async-to-LDS path
// (ASYNCcnt); outputs staged in LDS for coalesced streaming.

#define B_        16
#define L_        131072
#define DIM_      9
#define SR_       22050.0f
#define SINE_AMP_ 0.1f
#define NOISE_STD_ 0.003f

#define CHUNK   1024            // time steps per block in k_partial / k_sine
#define NCHUNK  (L_ / CHUNK)    // 128 chunks per channel
#define TILES   (CHUNK / 256)   // 4 WMMA tiles per wave per chunk

typedef __attribute__((ext_vector_type(2))) float v2f;
typedef __attribute__((ext_vector_type(8))) float v8f;

#define AS1 __attribute__((address_space(1)))
#define AS3 __attribute__((address_space(3)))

#if defined(__gfx1250__) && __has_builtin(__builtin_amdgcn_global_load_async_to_lds_b32)
#define HAVE_ASYNC_LDS 1
#else
#define HAVE_ASYNC_LDS 0
#endif

__device__ __forceinline__ float fracf_(float x) { return x - floorf(x); }

// ---------------------------------------------------------------------------
// Kernel 1: per-(b,chunk) block computes per-harmonic chunk sums of
// frac(f0*k/SR).  ws[(b*9+d)*NCHUNK + chunk] = frac(sum).
// wave32 shfl_xor butterfly reduction, single barrier.
// ---------------------------------------------------------------------------
__global__ __launch_bounds__(256)
void k_partial(const float* __restrict__ f0, float* __restrict__ ws) {
  __shared__ float wred[8][DIM_];
  const int chunk = blockIdx.x, b = blockIdx.y, tid = threadIdx.x;
  const float* fp = f0 + (size_t)b * L_ + (size_t)chunk * CHUNK;

  float loc[DIM_];
#pragma unroll
  for (int d = 0; d < DIM_; ++d) loc[d] = 0.f;

  for (int i = tid; i < CHUNK; i += 256) {
    float f = fp[i];                       // coalesced; L2-resident on reuse
#pragma unroll
    for (int d = 0; d < DIM_; ++d)
      loc[d] += fracf_(f * ((float)(d + 1) / SR_));
  }

#pragma unroll
  for (int d = 0; d < DIM_; ++d) {
    float v = loc[d];
#pragma unroll
    for (int m = 16; m > 0; m >>= 1) v += __shfl_xor(v, m, 32);
    loc[d] = v;
  }
  const int wv = tid >> 5, ln = tid & 31;
  if (ln == 0) {
#pragma unroll
    for (int d = 0; d < DIM_; ++d) wred[wv][d] = loc[d];
  }
  __syncthreads();
  if (tid < DIM_) {
    float s = 0.f;
#pragma unroll
    for (int w = 0; w < 8; ++w) s += wred[w][tid];
    ws[((size_t)(b * DIM_ + tid)) * NCHUNK + chunk] = fracf_(s);
  }
}

// ---------------------------------------------------------------------------
// Kernel 2: per-channel exclusive scan of the NCHUNK partials (in place).
// Everything frac'd: sin(2*pi*x) is 1-periodic so dropping integers is exact.
// ---------------------------------------------------------------------------
__global__ __launch_bounds__(NCHUNK)
void k_scan(float* __restrict__ ws) {
  __shared__ float sc[NCHUNK];
  const int ch = blockIdx.x, j = threadIdx.x;
  float v = ws[(size_t)ch * NCHUNK + j];
  float acc = v;
  sc[j] = v;
  __syncthreads();
  for (int off = 1; off < NCHUNK; off <<= 1) {
    float t = (j >= off) ? sc[j - off] : 0.f;
    __syncthreads();
    acc += t;
    sc[j] = acc;
    __syncthreads();
  }
  ws[(size_t)ch * NCHUNK + j] = fracf_(acc - v);   // exclusive base, in [0,1)
}

// ---------------------------------------------------------------------------
// Kernel 3: 9 waves per block, wave d owns harmonic d for one 1024-step chunk.
// WMMA prefix scan: Y = T * X, T = 16x16 lower-triangular ones split into four
// K=4 chunks (A fragments are lane-computed constants); X[k][n] = rad[n*16+k].
// D layout: lane l (n=l&15, h=l>>4), VGPR v holds Y[v+8h][n] -> each lane
// produces 8 consecutive time steps. Carries via __shfl of column sums.
// ---------------------------------------------------------------------------
__global__ __launch_bounds__(288)
void k_sine(const float* __restrict__ f0, const float* __restrict__ rand_ini,
            const float* __restrict__ noise_z, const float* __restrict__ ws,
            float* __restrict__ out) {
  __shared__ float f0s[CHUNK];            // 4 KB
  __shared__ float sine_s[CHUNK * DIM_];  // 36 KB, [t][d] for coalesced flush

  const int chunk = blockIdx.x, b = blockIdx.y, tid = threadIdx.x;
  const int wave = tid >> 5, lane = tid & 31;
  const size_t t0 = (size_t)chunk * CHUNK;

  const size_t S = (size_t)B_ * L_ * DIM_;
  const size_t U = (size_t)B_ * L_;
  const size_t gbase = ((size_t)b * L_ + t0) * DIM_;

  // Warm L2/L0 for the noise tile consumed in the flush phase:
  // one 128B line per thread, 288 * 128B == CHUNK*DIM_*4B exactly.
  __builtin_prefetch((const void*)(noise_z + gbase + (size_t)tid * 32), 0, 1);

  const float* fp = f0 + (size_t)b * L_ + t0;
#if HAVE_ASYNC_LDS
  for (int i = tid; i < CHUNK; i += 288)
    __builtin_amdgcn_global_load_async_to_lds_b32(
        (AS1 int*)(fp + i), (AS3 int*)&f0s[i], 0, 0);
#if __has_builtin(__builtin_amdgcn_s_wait_asynccnt)
  __builtin_amdgcn_s_wait_asynccnt(0);
#else
  asm volatile("s_wait_asynccnt 0" ::: "memory");
#endif
#else
  for (int i = tid; i < CHUNK; i += 288) f0s[i] = fp[i];
#endif
  __syncthreads();

  // --- per-wave WMMA scan (wave == harmonic index d, all 9 waves active) ---
  const int d = wave;
  const float scale = (float)(d + 1) / SR_;
  const int ch = b * DIM_ + d;
  float carry = fracf_(ws[(size_t)ch * NCHUNK + chunk] + rand_ini[ch]);
  const int n = lane & 15, h = lane >> 4;   // n = column / A-row, h = half

  for (int tt = 0; tt < TILES; ++tt) {
    const int tb = tt * 256;
    v8f acc = {0.f, 0.f, 0.f, 0.f, 0.f, 0.f, 0.f, 0.f};
#pragma unroll
    for (int c = 0; c < 4; ++c) {
      const int k0 = 4 * c + 2 * h;        // this lane's two K indices
      v2f a, bb;
      a.x = (k0 <= n) ? 1.f : 0.f;         // T[n][k0]   (lower-triangular ones)
      a.y = (k0 + 1 <= n) ? 1.f : 0.f;     // T[n][k0+1]
      const int j0 = (n << 4) + k0;        // X[k][n] = x[n*16+k]
      bb.x = fracf_(f0s[tb + j0] * scale);
      bb.y = fracf_(f0s[tb + j0 + 1] * scale);
      acc = __builtin_amdgcn_wmma_f32_16x16x4_f32(
          /*neg_a=*/false, a, /*neg_b=*/false, bb,
          /*c_mod=*/(short)0, acc, /*reuse_a=*/false, /*reuse_b=*/false);
    }
    // Column offsets: colsum[m] = Y[15][m] lives in VGPR7 of lane 16+m.
    float c7 = acc[7];
    float off = 0.f, tot = 0.f;
#pragma unroll
    for (int mm = 0; mm < 16; ++mm) {
      float v = __shfl(c7, 16 + mm, 32);
      tot += v;
      if (mm < n) off += v;
    }
    const float basep = off + carry;
#pragma unroll
    for (int v = 0; v < 8; ++v) {
      const int i = v + 8 * h;             // D-matrix row
      const int tl = tb + (n << 4) + i;    // time step inside chunk
      float p = fracf_(acc[v] + basep);    // phase in [0,1)
      sine_s[tl * DIM_ + d] = SINE_AMP_ * __sinf(6.28318530718f * p);
    }
    carry = fracf_(carry + tot);
  }
  __syncthreads();

  // --- coalesced streaming flush: sine, noise (contiguous, d fastest), uv ---
  float* sine_o = out;
  float* uv_o = out + S;
  float* noise_o = out + S + U;

  for (int idx = tid; idx < CHUNK * DIM_; idx += 288) {
    const int tl = idx / DIM_;
    const float f = f0s[tl];
    const float uv = (f > 0.f) ? 1.f : 0.f;
    const float amp = uv * NOISE_STD_ + (1.f - uv) * (SINE_AMP_ / 3.f);
    const float nz = amp * noise_z[gbase + idx];
    noise_o[gbase + idx] = nz;
    sine_o[gbase + idx] = sine_s[idx] * uv + nz;
  }
  for (int i = tid; i < CHUNK; i += 288)
    uv_o[(size_t)b * L_ + t0 + i] = (f0s[i] > 0.f) ? 1.f : 0.f;
}

// ---------------------------------------------------------------------------
extern "C" void kernel_launch(void* const* d_in, const int* in_sizes, int n_in,
                              void* d_out, int out_size, void* d_ws,
                              size_t ws_size, hipStream_t stream) {
  const float* f0       = (const float*)d_in[0];  // (B, L, 1)
  const float* rand_ini = (const float*)d_in[1];  // (B, DIM)
  const float* noise_z  = (const float*)d_in[2];  // (B, L, DIM)
  float* out = (float*)d_out;                     // sine | uv | noise, flat
  float* ws  = (float*)d_ws;                      // 144*128 floats = 72 KB

  dim3 grid(NCHUNK, B_);
  k_partial<<<grid, 256, 0, stream>>>(f0, ws);
  k_scan<<<B_ * DIM_, NCHUNK, 0, stream>>>(ws);
  k_sine<<<grid, 288, 0, stream>>>(f0, rand_ini, noise_z, ws, out);
}